// RoiAlign_84516366450939
// MI455X (gfx1250) — compile-verified
//
#include <hip/hip_runtime.h>
#include <stdint.h>

// ---------------------------------------------------------------------------
// ROI Align, MI455X (gfx1250, wave32).
//  Pass 1: NCHW -> NHWC transpose into d_ws (coalesced both ways via LDS tile).
//  Pass 2: one block per ROI. All 49x16 bilinear corner indices/weights are
//  precomputed into LDS, then the 16 x 1KB corner channel-vectors per bin are
//  staged into LDS with CDNA5 async global->LDS copies (ASYNCcnt), DOUBLE-
//  BUFFERED so bin i+1's gather overlaps bin i's FMA reduction. Results land
//  in an LDS (c*49+bin) slab and stream out with fully coalesced stores.
// Roofline: ~0.5 GFLOP vs ~175MB unique HBM traffic (x fits in 192MB L2), so
// this is gather-bandwidth bound; WMMA has no non-redundant formulation here.
// ---------------------------------------------------------------------------

#define AS1 __attribute__((address_space(1)))
#define AS3 __attribute__((address_space(3)))

typedef int v4i __attribute__((ext_vector_type(4)));

__device__ __forceinline__ void async_load_b128(const float* gptr, float* lptr) {
#if __has_builtin(__builtin_amdgcn_global_load_async_to_lds_b128)
  __builtin_amdgcn_global_load_async_to_lds_b128((AS1 v4i*)gptr, (AS3 v4i*)lptr,
                                                 /*offset=*/0, /*cpol=*/0);
#else
  unsigned lds_off = (unsigned)(uintptr_t)(AS3 char*)lptr;
  asm volatile("global_load_async_to_lds_b128 %0, %1, off"
               :: "v"(lds_off), "v"(gptr) : "memory");
#endif
}

template <int N>
__device__ __forceinline__ void wait_async() {
#if __has_builtin(__builtin_amdgcn_s_wait_asynccnt)
  __builtin_amdgcn_s_wait_asynccnt(N);
#else
  asm volatile("s_wait_asynccnt %0" :: "i"(N) : "memory");
#endif
}

// Problem constants (from reference setup_inputs)
constexpr int BATCH = 8;
constexpr int CH    = 256;
constexpr int H_    = 100;
constexpr int W_    = 152;
constexpr int HW    = H_ * W_;          // 15200 = 32 * 475
constexpr int PH    = 7;
constexpr int PW    = 7;
constexpr int NBIN  = PH * PW;          // 49
constexpr float SCALE = 1.0f / 16.0f;

// ---------------------------------------------------------------------------
// Pass 1: NCHW -> NHWC transpose (per batch: [C, HW] -> [HW, C]), LDS-tiled.
// ---------------------------------------------------------------------------
__global__ __launch_bounds__(256)
void nchw_to_nhwc(const float* __restrict__ x, float* __restrict__ xt) {
  __shared__ float tile[32][33];
  const int p0 = blockIdx.x * 32;     // position (h*W+w) tile
  const int c0 = blockIdx.y * 32;     // channel tile
  const int b  = blockIdx.z;
  const int tx = threadIdx.x;         // 0..31
  const int ty = threadIdx.y;         // 0..7

  const float* src = x + (size_t)b * CH * HW;
#pragma unroll
  for (int j = 0; j < 32; j += 8)
    tile[ty + j][tx] = src[(size_t)(c0 + ty + j) * HW + (p0 + tx)];
  __syncthreads();

  float* dst = xt + (size_t)b * HW * CH;
#pragma unroll
  for (int j = 0; j < 32; j += 8)
    dst[(size_t)(p0 + ty + j) * CH + (c0 + tx)] = tile[tx][ty + j];
}

// ---------------------------------------------------------------------------
// Pass 2 helpers
// ---------------------------------------------------------------------------
__device__ __forceinline__ void stage_bin(const float* __restrict__ xt,
                                          const int* __restrict__ sp_bin,
                                          float* __restrict__ lbuf, int tid) {
  // 16 corners x 1KB: 256 threads x 4 async B128 copies (4 ASYNCcnt/wave32).
  const int slot   = tid >> 4;                 // 0..15 : corner
  const int lane16 = tid & 15;                 // 0..15
  const float* gbase = xt + (size_t)sp_bin[slot] * CH;
  float*       lbase = lbuf + slot * CH;
#pragma unroll
  for (int k = 0; k < 4; ++k) {
    const int ch = (lane16 + k * 16) * 4;      // 16B-granular chunk
    async_load_b128(gbase + ch, lbase + ch);
  }
}

// ---------------------------------------------------------------------------
// Pass 2: one block per ROI, double-buffered async staging.
// ---------------------------------------------------------------------------
__global__ __launch_bounds__(256)
void roi_align_nhwc(const float* __restrict__ xt, const float* __restrict__ rois,
                    float* __restrict__ out) {
  __shared__ float stage[2][16 * CH];    // 2 x 16 KB double-buffered corners
  __shared__ float acc[CH * NBIN];       // 49 KB result slab, (c*49 + bin)
  __shared__ int   sp_all[NBIN * 16];    // corner pixel index (b*H+y)*W + x
  __shared__ float sw_all[NBIN * 16];    // corner weight (valid * 0.25 folded)

  const int r   = blockIdx.x;
  const int tid = threadIdx.x;

  const float rb  = rois[r * 5 + 0];
  const float sx1 = rois[r * 5 + 1] * SCALE;
  const float sy1 = rois[r * 5 + 2] * SCALE;
  const float sx2 = rois[r * 5 + 3] * SCALE;
  const float sy2 = rois[r * 5 + 4] * SCALE;
  const int   b   = (int)rb;
  const float bin_w = fmaxf(sx2 - sx1, 1.0f) * (1.0f / PW);
  const float bin_h = fmaxf(sy2 - sy1, 1.0f) * (1.0f / PH);

  // Precompute all 49 bins x 4 samples (196 threads, 4 corners each).
  if (tid < NBIN * 4) {
    const int bin = tid >> 2;
    const int smp = tid & 3;
    const int ph = bin / PW, pw = bin % PW;
    const int iy = smp >> 1, ix = smp & 1;
    const float y = sy1 + ((float)ph + ((float)iy + 0.5f) * 0.5f) * bin_h;
    const float x = sx1 + ((float)pw + ((float)ix + 0.5f) * 0.5f) * bin_w;
    const bool valid = (y >= -1.0f) && (y <= (float)H_) &&
                       (x >= -1.0f) && (x <= (float)W_);
    const float yc = fminf(fmaxf(y, 0.0f), (float)(H_ - 1));
    const float xc = fminf(fmaxf(x, 0.0f), (float)(W_ - 1));
    const int y0 = (int)floorf(yc), x0 = (int)floorf(xc);
    const int y1 = min(y0 + 1, H_ - 1), x1 = min(x0 + 1, W_ - 1);
    const float ly = yc - (float)y0, lx = xc - (float)x0;
    const float hy = 1.0f - ly,      hx = 1.0f - lx;
    const float vs = valid ? 0.25f : 0.0f;     // fold sample-average + mask
    const int base = b * HW;
    const int s = bin * 16 + smp * 4;
    sp_all[s + 0] = base + y0 * W_ + x0;  sw_all[s + 0] = hy * hx * vs;
    sp_all[s + 1] = base + y0 * W_ + x1;  sw_all[s + 1] = hy * lx * vs;
    sp_all[s + 2] = base + y1 * W_ + x0;  sw_all[s + 2] = ly * hx * vs;
    sp_all[s + 3] = base + y1 * W_ + x1;  sw_all[s + 3] = ly * lx * vs;
  }
  __syncthreads();

  // Prologue: stage bin 0 into buffer 0.
  stage_bin(xt, &sp_all[0], stage[0], tid);

  for (int bin = 0; bin < NBIN; ++bin) {
    const int cur = bin & 1;
    if (bin + 1 < NBIN) {
      // Issue next bin's gather into the other buffer, then wait only for the
      // current bin's copies (async copies complete in order; <=4 leaves the
      // 4 just-issued in flight).
      stage_bin(xt, &sp_all[(bin + 1) * 16], stage[cur ^ 1], tid);
      wait_async<4>();
    } else {
      wait_async<0>();
    }
    __syncthreads();   // all waves' copies for `bin` have landed in LDS

    // Weighted 16-term reduction for this thread's channel.
    const float* sw = &sw_all[bin * 16];
    const float* sg = stage[cur];
    float a = 0.0f;
#pragma unroll
    for (int k = 0; k < 16; ++k)
      a = fmaf(sw[k], sg[k * CH + tid], a);
    acc[tid * NBIN + bin] = a;   // stride 49 (odd) -> conflict-free, 64 banks
    __syncthreads();   // everyone done reading stage[cur] before it's reused
  }

  // Fully coalesced 49KB store of the (C, 7, 7) slab for this ROI.
  float* o = out + (size_t)r * (CH * NBIN);
  for (int i = tid; i < CH * NBIN; i += 256)
    o[i] = acc[i];
}

// ---------------------------------------------------------------------------
// Fallback: direct NCHW gather (used only if d_ws is too small for NHWC copy)
// ---------------------------------------------------------------------------
__global__ __launch_bounds__(256)
void roi_align_nchw_fallback(const float* __restrict__ x,
                             const float* __restrict__ rois,
                             float* __restrict__ out, int total) {
  const int idx = blockIdx.x * blockDim.x + threadIdx.x;
  if (idx >= total) return;
  const int bin = idx % NBIN;
  const int t   = idx / NBIN;
  const int c   = t % CH;
  const int r   = t / CH;
  const int ph = bin / PW, pw = bin % PW;

  const int   b   = (int)rois[r * 5 + 0];
  const float sx1 = rois[r * 5 + 1] * SCALE;
  const float sy1 = rois[r * 5 + 2] * SCALE;
  const float sx2 = rois[r * 5 + 3] * SCALE;
  const float sy2 = rois[r * 5 + 4] * SCALE;
  const float bin_w = fmaxf(sx2 - sx1, 1.0f) * (1.0f / PW);
  const float bin_h = fmaxf(sy2 - sy1, 1.0f) * (1.0f / PH);

  const float* xp = x + ((size_t)b * CH + c) * HW;
  float a = 0.0f;
#pragma unroll
  for (int s = 0; s < 4; ++s) {
    const int iy = s >> 1, ix = s & 1;
    const float y  = sy1 + ((float)ph + ((float)iy + 0.5f) * 0.5f) * bin_h;
    const float xf = sx1 + ((float)pw + ((float)ix + 0.5f) * 0.5f) * bin_w;
    const bool valid = (y >= -1.0f) && (y <= (float)H_) &&
                       (xf >= -1.0f) && (xf <= (float)W_);
    const float yc = fminf(fmaxf(y, 0.0f), (float)(H_ - 1));
    const float xc = fminf(fmaxf(xf, 0.0f), (float)(W_ - 1));
    const int y0 = (int)floorf(yc), x0 = (int)floorf(xc);
    const int y1 = min(y0 + 1, H_ - 1), x1 = min(x0 + 1, W_ - 1);
    const float ly = yc - (float)y0, lx = xc - (float)x0;
    const float hy = 1.0f - ly,      hx = 1.0f - lx;
    float v = hy * hx * xp[y0 * W_ + x0] + hy * lx * xp[y0 * W_ + x1] +
              ly * hx * xp[y1 * W_ + x0] + ly * lx * xp[y1 * W_ + x1];
    a += valid ? v : 0.0f;
  }
  out[idx] = a * 0.25f;
}

// ---------------------------------------------------------------------------
extern "C" void kernel_launch(void* const* d_in, const int* in_sizes, int n_in,
                              void* d_out, int out_size, void* d_ws, size_t ws_size,
                              hipStream_t stream) {
  const float* x    = (const float*)d_in[0];
  const float* rois = (const float*)d_in[1];
  float*       out  = (float*)d_out;
  const int R = in_sizes[1] / 5;

  const size_t nhwc_bytes = (size_t)BATCH * HW * CH * sizeof(float);
  if (ws_size >= nhwc_bytes) {
    float* xt = (float*)d_ws;
    nchw_to_nhwc<<<dim3(HW / 32, CH / 32, BATCH), dim3(32, 8), 0, stream>>>(x, xt);
    roi_align_nhwc<<<R, 256, 0, stream>>>(xt, rois, out);
  } else {
    const int total = R * CH * NBIN;
    roi_align_nchw_fallback<<<(total + 255) / 256, 256, 0, stream>>>(x, rois, out, total);
  }
}